// SequenceToPairTE_32031866094096
// MI455X (gfx1250) — compile-verified
//
#include <hip/hip_runtime.h>
#include <hip/hip_bf16.h>

// Sizes from the reference
#define BB    2
#define LL    512
#define DD    1024
#define INNER 32
#define PAIR  128

typedef __attribute__((ext_vector_type(2))) float v2f;
typedef __attribute__((ext_vector_type(8))) float v8f;

// ---------------------------------------------------------------------------
// Kernel 1: LayerNorm over D + projection to 2*INNER, split into q / k.
// One block per (b,l) row. 256 threads = 8 waves (wave32).
// ---------------------------------------------------------------------------
__global__ __launch_bounds__(256) void ln_proj_kernel(
    const float* __restrict__ x,      // B*L*D
    const float* __restrict__ gamma,  // D
    const float* __restrict__ beta,   // D
    const float* __restrict__ Wproj,  // (2*INNER, D) row-major
    const float* __restrict__ bproj,  // 2*INNER
    float* __restrict__ q,            // B*L x INNER
    float* __restrict__ k)            // B*L x INNER
{
    const int row = blockIdx.x;                 // b*L + l
    const float* xr = x + (size_t)row * DD;

    __shared__ float sn[DD];
    __shared__ float ws1[8], ws2[8];

    const int tid  = threadIdx.x;
    const int wid  = tid >> 5;
    const int lane = tid & 31;

    // --- mean / var (each thread owns 4 contiguous floats) ---
    float4 v = ((const float4*)xr)[tid];
    float s  = v.x + v.y + v.z + v.w;
    float s2 = v.x * v.x + v.y * v.y + v.z * v.z + v.w * v.w;
    #pragma unroll
    for (int off = 16; off > 0; off >>= 1) {
        s  += __shfl_xor(s,  off, 32);
        s2 += __shfl_xor(s2, off, 32);
    }
    if (lane == 0) { ws1[wid] = s; ws2[wid] = s2; }
    __syncthreads();
    if (tid == 0) {
        float a = 0.f, b2 = 0.f;
        #pragma unroll
        for (int w = 0; w < 8; ++w) { a += ws1[w]; b2 += ws2[w]; }
        const float mu  = a * (1.0f / DD);
        const float var = b2 * (1.0f / DD) - mu * mu;
        ws1[0] = mu;
        ws2[0] = rsqrtf(var + 1e-5f);
    }
    __syncthreads();
    const float mu = ws1[0];
    const float rs = ws2[0];

    // --- normalized row into LDS ---
    {
        float4 g = ((const float4*)gamma)[tid];
        float4 bt = ((const float4*)beta)[tid];
        float4 o;
        o.x = (v.x - mu) * rs * g.x + bt.x;
        o.y = (v.y - mu) * rs * g.y + bt.y;
        o.z = (v.z - mu) * rs * g.z + bt.z;
        o.w = (v.w - mu) * rs * g.w + bt.w;
        ((float4*)sn)[tid] = o;
    }
    __syncthreads();

    // --- projection: wave w computes outputs e = w*8 .. w*8+7 ---
    #pragma unroll
    for (int ee = 0; ee < 8; ++ee) {
        const int e = wid * 8 + ee;
        const float* wr = Wproj + (size_t)e * DD;
        float acc = 0.f;
        #pragma unroll 8
        for (int c = 0; c < DD; c += 32)
            acc = fmaf(wr[c + lane], sn[c + lane], acc);
        #pragma unroll
        for (int off = 16; off > 0; off >>= 1)
            acc += __shfl_xor(acc, off, 32);
        if (lane == 0) {
            const float val = acc + bproj[e];
            if (e < INNER) q[(size_t)row * INNER + e] = val;
            else           k[(size_t)row * INNER + (e - INNER)] = val;
        }
    }
}

// ---------------------------------------------------------------------------
// Kernel 2: pairwise output via folded GEMM.
//   W'[p,e]  = Wo[p,e]*k[bi,e] + Wo[p,e+32]
//   bias'[p] = bo[p] - sum_e Wo[p,e+32]*k[bi,e]
//   out[bi,j,p] = sum_e q[b,j,e]*W'[p,e] + bias'[p]
// One block per (b,i): a 512x128 (K=32) GEMM with V_WMMA_F32_16X16X4_F32.
// 8 waves; wave w handles j-tiles w*4..w*4+3, each against all 8 p-tiles.
// ---------------------------------------------------------------------------
#define WSTRIDE 34   // pad 32 -> 34: float2 LDS loads stay 8B-aligned, no bank conflicts

__global__ __launch_bounds__(256) void pair_wmma_kernel(
    const float* __restrict__ q,    // B*L x INNER
    const float* __restrict__ k,    // B*L x INNER
    const float* __restrict__ Wo,   // (PAIR, 2*INNER) row-major
    const float* __restrict__ bo,   // PAIR
    float* __restrict__ out)        // B x L x L x PAIR
{
    const int bi = blockIdx.x;      // b*L + i
    const int b  = bi >> 9;

    __shared__ float Wl[PAIR * WSTRIDE];
    __shared__ float bias2[PAIR];
    __shared__ float ks[INNER];

    const int tid = threadIdx.x;

    if (tid < INNER) ks[tid] = k[(size_t)bi * INNER + tid];
    __syncthreads();

    // Build W' (128x32) in LDS
    for (int idx = tid; idx < PAIR * INNER; idx += 256) {
        const int p = idx >> 5;
        const int e = idx & 31;
        Wl[p * WSTRIDE + e] = Wo[p * 64 + e] * ks[e] + Wo[p * 64 + 32 + e];
    }
    // bias'[p]
    if (tid < PAIR) {
        float acc = bo[tid];
        const float* wd = Wo + tid * 64 + 32;
        #pragma unroll 8
        for (int e = 0; e < INNER; ++e) acc -= wd[e] * ks[e];
        bias2[tid] = acc;
    }
    __syncthreads();

    const int wid   = tid >> 5;
    const int lane  = tid & 31;
    const int mrow  = lane & 15;          // M (or N) index within tile
    const int khalf = (lane >> 4) * 2;    // K sub-pair selector per ISA layout

    const float* qb   = q + (size_t)b * LL * INNER;
    float*       outb = out + (size_t)bi * LL * PAIR;

    #pragma unroll 1
    for (int jj = 0; jj < 4; ++jj) {
        const int jt = wid * 4 + jj;

        // A fragments: q rows jt*16..jt*16+15, all K=32 as 8 k-steps of 4.
        // lane layout: M = lane%16 ; VGPR pair holds K = s*4 + khalf + {0,1}
        v2f a[8];
        const float* qrow = qb + (size_t)(jt * 16 + mrow) * INNER;
        #pragma unroll
        for (int s = 0; s < 8; ++s)
            a[s] = *(const v2f*)(qrow + s * 4 + khalf);

        #pragma unroll 1
        for (int pt = 0; pt < 8; ++pt) {
            const float bias = bias2[pt * 16 + mrow];
            v8f c = { bias, bias, bias, bias, bias, bias, bias, bias };

            // B fragments from LDS: N = lane%16 (= p within tile),
            // K = s*4 + khalf + {0,1}
            const float* wrow = &Wl[(pt * 16 + mrow) * WSTRIDE];
            #pragma unroll
            for (int s = 0; s < 8; ++s) {
                v2f bb = *(const v2f*)(wrow + s * 4 + khalf);
                c = __builtin_amdgcn_wmma_f32_16x16x4_f32(
                        /*neg_a=*/false, a[s],
                        /*neg_b=*/false, bb,
                        /*c_mod=*/(short)0, c,
                        /*reuse_a=*/false, /*reuse_b=*/false);
            }

            // Store: C VGPR v -> row j = jt*16 + v + 8*(lane/16), col p = pt*16 + lane%16
            float* op = outb + (size_t)(jt * 16 + (lane >> 4) * 8) * PAIR
                             + pt * 16 + mrow;
            #pragma unroll
            for (int v = 0; v < 8; ++v)
                op[(size_t)v * PAIR] = c[v];
        }
    }
}

// ---------------------------------------------------------------------------
extern "C" void kernel_launch(void* const* d_in, const int* in_sizes, int n_in,
                              void* d_out, int out_size, void* d_ws, size_t ws_size,
                              hipStream_t stream) {
    const float* seq    = (const float*)d_in[0]; // B*L*D
    const float* gamma  = (const float*)d_in[1]; // D
    const float* beta   = (const float*)d_in[2]; // D
    const float* Wproj  = (const float*)d_in[3]; // (64, D)
    const float* bproj  = (const float*)d_in[4]; // 64
    const float* Wo     = (const float*)d_in[5]; // (128, 64)
    const float* bo     = (const float*)d_in[6]; // 128
    float* out = (float*)d_out;

    float* qws = (float*)d_ws;                        // B*L*INNER floats
    float* kws = qws + (size_t)BB * LL * INNER;       // B*L*INNER floats

    ln_proj_kernel<<<BB * LL, 256, 0, stream>>>(seq, gamma, beta, Wproj, bproj,
                                                qws, kws);
    pair_wmma_kernel<<<BB * LL, 256, 0, stream>>>(qws, kws, Wo, bo, out);
}